// BoundaryConsistencyLoss_58875411693879
// MI455X (gfx1250) — compile-verified
//
#include <hip/hip_runtime.h>
#include <stdint.h>

#define B_ 64
#define H_ 512
#define W_ 512
#define TILE_H 16
#define NWAVE (W_ / 32)          // 16 ballot words per row
#define LINE_T 300u

typedef unsigned int u32x4 __attribute__((ext_vector_type(4)));
typedef int i32x8 __attribute__((ext_vector_type(8)));
typedef int i32x4 __attribute__((ext_vector_type(4)));

__device__ __forceinline__ unsigned ballot32(bool p) {
  return __builtin_amdgcn_ballot_w32(p);
}

// ---------------------------------------------------------------------------
// Pass 1: per (input, image, 16-row band): TDM-stage halo tile into LDS,
// then 3 barrier-separated phases: vertical or/and -> horizontal or/and +
// ballot -> row-filter + column counts + packed bit words.
// ---------------------------------------------------------------------------
__global__ __launch_bounds__(512)
void boundary_rows_kernel(const float* __restrict__ seg,
                          const float* __restrict__ gt,
                          unsigned* __restrict__ bits,
                          unsigned* __restrict__ colsum)
{
  __shared__ float tile[(TILE_H + 2) * W_];          // 36,864 B (TDM target)
  __shared__ unsigned char hxA[TILE_H * W_];         //  8,192 B: {AND<<1|OR}
  __shared__ unsigned wsum[TILE_H * NWAVE];          //  1,024 B

  const int band = blockIdx.x;
  const int img  = blockIdx.y;
  const int z    = blockIdx.z;
  const float* src = z ? gt : seg;
  unsigned* bitsbase = bits + (size_t)z * ((size_t)B_ * H_ * NWAVE);
  unsigned* csum     = colsum + ((size_t)z * B_ + img) * W_;

  const int h0 = band * TILE_H;
  const int g0 = (h0 > 0) ? (h0 - 1) : 0;
  const int g1 = (h0 + TILE_H < H_) ? (h0 + TILE_H) : (H_ - 1);
  const int nrows = g1 - g0 + 1;                     // 17 or 18

  // ---- Tensor Data Mover: async DMA (nrows x 512) f32 tile global->LDS ----
  if ((threadIdx.x >> 5) == 0) {                     // one wave issues the DMA
    unsigned long long gaddr =
        (unsigned long long)(uintptr_t)(src + ((size_t)img * H_ + (size_t)g0) * W_);
    unsigned ldsoff = (unsigned)(uintptr_t)(void*)&tile[0];
    // D# group 0 (ISA 8.3): count=1 | lds_addr | global_addr[56:0] | type=2
    u32x4 d0;
    d0.x = 1u;
    d0.y = ldsoff;
    d0.z = (unsigned)(gaddr & 0xFFFFFFFFull);
    d0.w = (unsigned)((gaddr >> 32) & 0x1FFFFFFull) | (2u << 30);
    // D# group 1 (ISA 8.4): data_size=4B, dims/strides in elements
    i32x8 d1;
    d1[0] = 2 << 16;                  // wg_mask=0 (not in cluster), data_size=2 (4B)
    d1[1] = (W_ & 0xFFFF) << 16;      // tensor_dim0[15:0] @ bits 63:48
    d1[2] = (nrows & 0xFFFF) << 16;   // tensor_dim0[31:16]=0, tensor_dim1[15:0]
    d1[3] = (W_ & 0xFFFF) << 16;      // tensor_dim1[31:16]=0, tile_dim0=512
    d1[4] = nrows & 0xFFFF;           // tile_dim1=nrows, tile_dim2=0
    d1[5] = W_;                       // tensor_dim0_stride low32
    d1[6] = 0;
    d1[7] = 0;
    i32x4 zz = {0, 0, 0, 0};          // groups 2/3 unused (2-D tensor)
#if __clang_major__ >= 23
    i32x8 z8 = {0, 0, 0, 0, 0, 0, 0, 0};
    __builtin_amdgcn_tensor_load_to_lds(d0, d1, zz, zz, z8, 0);
#else
    __builtin_amdgcn_tensor_load_to_lds(d0, d1, zz, zz, 0);
#endif
    __builtin_amdgcn_s_wait_tensorcnt(0);
  }
  __syncthreads();                                   // tile ready for all waves

  const int c    = threadIdx.x;                      // column 0..511
  const int lane = c & 31;
  const int wid  = c >> 5;
  const int cl   = (c == 0) ? 0 : c - 1;
  const int cr   = (c == W_ - 1) ? (W_ - 1) : c + 1;

  // ---- Phase A: vertical OR/AND with a rolling 3-row register window ----
  {
    unsigned xm = (tile[(((h0 > 0) ? h0 - 1 : 0) - g0) * W_ + c] > 0.5f) ? 1u : 0u;
    unsigned xc = (tile[(h0 - g0) * W_ + c] > 0.5f) ? 1u : 0u;
    #pragma unroll
    for (int i = 0; i < TILE_H; ++i) {
      const int h  = h0 + i;
      const int rp = ((h < H_ - 1) ? h + 1 : H_ - 1) - g0;
      const unsigned xp = (tile[rp * W_ + c] > 0.5f) ? 1u : 0u;
      const unsigned vo = xm | xc | xp;
      const unsigned va = xm & xc & xp;
      hxA[i * W_ + c] = (unsigned char)(vo | (va << 1));
      xm = xc; xc = xp;
    }
  }
  __syncthreads();

  // ---- Phase B: horizontal OR/AND -> boundary bit -> wave32 ballot ----
  unsigned balArr[TILE_H];
  #pragma unroll
  for (int i = 0; i < TILE_H; ++i) {
    const unsigned L = hxA[i * W_ + cl];
    const unsigned C = hxA[i * W_ + c];
    const unsigned R = hxA[i * W_ + cr];
    // boundary = OR3x3 & !AND3x3 (binary data: max/min == or/and)
    const unsigned bbit = ((L | C | R) & 1u) & ((~((L & C & R) >> 1)) & 1u);
    const unsigned bal = ballot32(bbit != 0u);       // 32 cols -> 1 word
    balArr[i] = bal;
    if (lane == 0) wsum[i * NWAVE + wid] = (unsigned)__popc(bal);
  }
  __syncthreads();

  // ---- Phase C: row filter (shfl butterfly), column counts, bit words ----
  unsigned colcnt = 0;
  #pragma unroll
  for (int i = 0; i < TILE_H; ++i) {
    unsigned v = wsum[i * NWAVE + (lane & (NWAVE - 1))];
    v += __shfl_xor(v, 8, 32);
    v += __shfl_xor(v, 4, 32);
    v += __shfl_xor(v, 2, 32);
    v += __shfl_xor(v, 1, 32);                       // all lanes: row total
    const unsigned keep = (v < LINE_T) ? 1u : 0u;    // row-sum >= 300 -> zero
    colcnt += ((balArr[i] >> lane) & 1u) & keep;
    if (lane == 0)
      bitsbase[((size_t)img * H_ + (size_t)(h0 + i)) * NWAVE + wid] =
          keep ? balArr[i] : 0u;
  }
  atomicAdd(&csum[c], colcnt);
}

// ---------------------------------------------------------------------------
// Pass 2: pack (colsum >= 300) into 16 mask words per image per input.
// ---------------------------------------------------------------------------
__global__ __launch_bounds__(512)
void colhit_kernel(const unsigned* __restrict__ colsum,
                   unsigned* __restrict__ colhit)
{
  const int img = blockIdx.x, z = blockIdx.y;
  const int c = threadIdx.x, lane = c & 31, wid = c >> 5;
  const unsigned s = colsum[((size_t)z * B_ + img) * W_ + c];
  const unsigned m = ballot32(s >= LINE_T);
  if (lane == 0) colhit[((size_t)z * B_ + img) * NWAVE + wid] = m;
}

// ---------------------------------------------------------------------------
// Pass 3: mismatch popcount over packed words (reads ~4 MB, L2-resident).
// ---------------------------------------------------------------------------
__global__ __launch_bounds__(256)
void mismatch_kernel(const unsigned* __restrict__ bits,
                     const unsigned* __restrict__ colhit,
                     unsigned* __restrict__ count)
{
  __shared__ unsigned red[256];
  const unsigned WPI = (unsigned)B_ * H_ * NWAVE;        // 524288 words/input
  const unsigned idx = blockIdx.x * 256u + threadIdx.x;  // one word pair
  const unsigned img = idx >> 13;                        // / (H_*NWAVE)
  const unsigned wv  = idx & (NWAVE - 1);
  const unsigned a = bits[idx]       & ~colhit[img * NWAVE + wv];
  const unsigned b = bits[WPI + idx] & ~colhit[(unsigned)B_ * NWAVE + img * NWAVE + wv];
  red[threadIdx.x] = (unsigned)__popc(a ^ b);
  __syncthreads();
  for (int s = 128; s > 0; s >>= 1) {
    if ((int)threadIdx.x < s) red[threadIdx.x] += red[threadIdx.x + s];
    __syncthreads();
  }
  if (threadIdx.x == 0) atomicAdd(count, red[0]);
}

__global__ void finalize_kernel(const unsigned* __restrict__ count,
                                float* __restrict__ out)
{
  // boundary_mask = 2*b each side -> loss = 4 * mismatches / (B*H*W)
  out[0] = (float)((double)count[0] * (4.0 / (double)((size_t)B_ * H_ * W_)));
}

extern "C" void kernel_launch(void* const* d_in, const int* in_sizes, int n_in,
                              void* d_out, int out_size, void* d_ws, size_t ws_size,
                              hipStream_t stream)
{
  const float* seg = (const float*)d_in[0];
  const float* gt  = (const float*)d_in[1];

  unsigned* ws = (unsigned*)d_ws;
  const size_t WPI = (size_t)B_ * H_ * NWAVE;            // 524288
  unsigned* bits   = ws;                                 // 2*WPI words (4 MB)
  unsigned* colsum = bits + 2 * WPI;                     // 2*B_*W_ words
  unsigned* colhit = colsum + 2 * (size_t)B_ * W_;       // 2*B_*NWAVE words
  unsigned* count  = colhit + 2 * (size_t)B_ * NWAVE;    // 1 word

  // zero colsum + colhit + count (contiguous)
  hipMemsetAsync(colsum, 0,
                 (2 * (size_t)B_ * W_ + 2 * (size_t)B_ * NWAVE + 1) * sizeof(unsigned),
                 stream);

  boundary_rows_kernel<<<dim3(H_ / TILE_H, B_, 2), 512, 0, stream>>>(seg, gt, bits, colsum);
  colhit_kernel<<<dim3(B_, 2), W_, 0, stream>>>(colsum, colhit);
  mismatch_kernel<<<(unsigned)(WPI / 256), 256, 0, stream>>>(bits, colhit, count);
  finalize_kernel<<<1, 1, 0, stream>>>(count, (float*)d_out);
}